// GATModel_11510512353285
// MI455X (gfx1250) — compile-verified
//
#include <hip/hip_runtime.h>
#include <math.h>

#define NEG_SLOPE 0.2f

typedef __attribute__((ext_vector_type(2))) float v2f;
typedef __attribute__((ext_vector_type(8))) float v8f;

// ---------------------------------------------------------------- utilities

__global__ void fill_zero_kernel(float* __restrict__ p, long n) {
  long i = (long)blockIdx.x * blockDim.x + threadIdx.x;
  if (i < n) p[i] = 0.0f;
}

// order-preserving float <-> uint encoding for atomicMax on floats
__device__ __forceinline__ unsigned fenc(float f) {
  unsigned u = __float_as_uint(f);
  return (u & 0x80000000u) ? ~u : (u | 0x80000000u);
}
__device__ __forceinline__ float fdec(unsigned e) {
  return (e & 0x80000000u) ? __uint_as_float(e & 0x7fffffffu)
                           : __uint_as_float(~e);
}

// ------------------------------------------------- fp32 WMMA GEMM (16x16x4)
// C[M,Nc] = A[M,K] @ B[K,Nc], all row-major fp32.
// Requires M,Nc multiples of 16 and K multiple of 4 (true here: 50000,128/16,128).
// One wave32 computes one 16x16 tile of C via V_WMMA_F32_16X16X4_F32.
__global__ void wmma_gemm_f32_kernel(const float* __restrict__ A,
                                     const float* __restrict__ B,
                                     float* __restrict__ C,
                                     int M, int K, int Nc) {
  int wave = (int)((blockIdx.x * (long)blockDim.x + threadIdx.x) >> 5);
  int lane = threadIdx.x & 31;
  int colTiles = Nc >> 4;
  int nTiles = (M >> 4) * colTiles;
  if (wave >= nTiles) return;                 // wave-uniform: EXEC stays all-1s
  int rt = wave / colTiles;
  int ct = wave - rt * colTiles;
  int row0 = rt << 4, col0 = ct << 4;
  int hf = lane >> 4;                         // lane half: 0 -> K{0,1}, 1 -> K{2,3}
  int l  = lane & 15;                         // M index (A) / N index (B,C)

  const float* Arow = A + (long)(row0 + l) * K;
  v8f acc = {};
#pragma unroll 8
  for (int k0 = 0; k0 < K; k0 += 4) {
    int ka = k0 + (hf << 1);
    v2f av, bv;
    // A 16x4: VGPR0 = K=ka, VGPR1 = K=ka+1 (per ISA layout table)
    av.x = Arow[ka];
    av.y = Arow[ka + 1];
    // B 4x16: lanes 0-15 hold K=0,1 ; lanes 16-31 hold K=2,3 ; col = lane&15
    bv.x = B[(long)ka * Nc + col0 + l];
    bv.y = B[(long)(ka + 1) * Nc + col0 + l];
    acc = __builtin_amdgcn_wmma_f32_16x16x4_f32(
        /*neg_a=*/false, av, /*neg_b=*/false, bv,
        /*c_mod=*/(short)0, acc, /*reuse_a=*/false, /*reuse_b=*/false);
  }
  // C 16x16 f32: VGPR i -> row = i + 8*half, col = lane&15
#pragma unroll
  for (int i = 0; i < 8; ++i) {
    int r = row0 + hf * 8 + i;
    C[(long)r * Nc + col0 + l] = acc[i];
  }
}

// ------------------------------------------------ attention score per (n,h)
// a_s[n,h] = sum_c h[n,h,c]*att_src[h,c] ; a_d likewise
__global__ void att_kernel(const float* __restrict__ h,
                           const float* __restrict__ att_s,
                           const float* __restrict__ att_d,
                           float* __restrict__ as_, float* __restrict__ ad_,
                           int N, int heads, int ch) {
  long i = (long)blockIdx.x * blockDim.x + threadIdx.x;  // n*heads + hd
  if (i >= (long)N * heads) return;
  int hd = (int)(i % heads);
  const float* row = h + i * ch;   // h row-major [N, heads*ch]
  float ss = 0.f, sd = 0.f;
  for (int c = 0; c < ch; ++c) {
    float v = row[c];
    ss += v * att_s[hd * ch + c];
    sd += v * att_d[hd * ch + c];
  }
  as_[i] = ss;
  ad_[i] = sd;
}

// --------------------------------------------- pass 1: per-(dst,head) max e
__global__ void edge_max_kernel(const long long* __restrict__ ei, int E, int N,
                                const float* __restrict__ as_,
                                const float* __restrict__ ad_,
                                unsigned* __restrict__ m, int heads) {
  int e = blockIdx.x * blockDim.x + threadIdx.x;
  if (e >= E + N) return;
  int s, d;
  if (e < E) { s = (int)ei[e]; d = (int)ei[E + e]; }
  else       { s = d = e - E; }                    // appended self-loops
  for (int hd = 0; hd < heads; ++hd) {
    float x = as_[(long)s * heads + hd] + ad_[(long)d * heads + hd];
    x = (x >= 0.f) ? x : NEG_SLOPE * x;            // leaky_relu
    atomicMax(&m[(long)d * heads + hd], fenc(x));
  }
}

// ------------- pass 2: w = exp(e - max); accumulate denom and weighted feats
__global__ void edge_agg_kernel(const long long* __restrict__ ei, int E, int N,
                                const float* __restrict__ as_,
                                const float* __restrict__ ad_,
                                const unsigned* __restrict__ m,
                                float* __restrict__ ssum,
                                const float* __restrict__ h,
                                float* __restrict__ agg,
                                int heads, int ch) {
  long i = (long)blockIdx.x * blockDim.x + threadIdx.x;   // (edge, head)
  long tot = (long)(E + N) * heads;
  if (i >= tot) return;
  int e = (int)(i / heads);
  int hd = (int)(i - (long)e * heads);
  int s, d;
  if (e < E) { s = (int)ei[e]; d = (int)ei[E + e]; }
  else       { s = d = e - E; }
  float x = as_[(long)s * heads + hd] + ad_[(long)d * heads + hd];
  x = (x >= 0.f) ? x : NEG_SLOPE * x;
  float w = expf(x - fdec(m[(long)d * heads + hd]));
  atomicAdd(&ssum[(long)d * heads + hd], w);
  const float* hs = h + ((long)s * heads + hd) * ch;
  float* ag = agg + ((long)d * heads + hd) * ch;
  for (int c = 0; c < ch; ++c) atomicAdd(&ag[c], w * hs[c]);
}

// -------------------- normalize by softmax denom, add bias, optional ELU
__global__ void norm_kernel(float* __restrict__ agg, const float* __restrict__ ssum,
                            const float* __restrict__ bias, int N, int heads,
                            int ch, int do_elu) {
  long i = (long)blockIdx.x * blockDim.x + threadIdx.x;
  long tot = (long)N * heads * ch;
  if (i >= tot) return;
  int per = heads * ch;
  int cf = (int)(i % per);
  int hd = cf / ch;
  long n = i / per;
  float v = agg[i] / (ssum[n * heads + hd] + 1e-16f) + bias[cf];
  if (do_elu) v = (v > 0.f) ? v : expm1f(v);       // ELU(alpha=1)
  agg[i] = v;
}

// ---------------------------------------------------------------- launcher
extern "C" void kernel_launch(void* const* d_in, const int* in_sizes, int n_in,
                              void* d_out, int out_size, void* d_ws, size_t ws_size,
                              hipStream_t stream) {
  const float*     x   = (const float*)d_in[0];
  const long long* ei  = (const long long*)d_in[1];   // int64 [2,E]
  const float*     W1  = (const float*)d_in[2];       // [128,128]
  const float*     as1 = (const float*)d_in[3];       // [8,16]
  const float*     ad1 = (const float*)d_in[4];
  const float*     b1  = (const float*)d_in[5];       // [128]
  const float*     W2  = (const float*)d_in[6];       // [128,16]
  const float*     as2 = (const float*)d_in[7];       // [1,16]
  const float*     ad2 = (const float*)d_in[8];
  const float*     b2  = (const float*)d_in[9];       // [16]
  float* out = (float*)d_out;

  const int IN_CH = 128, HEADS = 8, HID = 16, OUTC = 16;
  const int N = in_sizes[0] / IN_CH;   // 50000
  const int E = in_sizes[1] / 2;       // 800000
  const int ET = E + N;                // edges incl. self loops

  // workspace carve-up (floats)
  float* ws = (float*)d_ws;
  float*    h1   = ws;                       // N*128
  float*    a1s  = h1  + (long)N * 128;      // N*8
  float*    a1d  = a1s + (long)N * 8;        // N*8
  unsigned* m1   = (unsigned*)(a1d + (long)N * 8);  // N*8
  float*    s1   = (float*)m1 + (long)N * 8; // N*8
  float*    agg1 = s1  + (long)N * 8;        // N*128 (becomes ELU'd layer-1 out)
  float*    h2   = agg1 + (long)N * 128;     // N*16
  float*    a2s  = h2  + (long)N * 16;       // N
  float*    a2d  = a2s + (long)N;            // N
  unsigned* m2   = (unsigned*)(a2d + (long)N); // N
  float*    s2   = (float*)m2 + (long)N;     // N

  const int TB = 256;
  #define GRID(n) ((int)(((long)(n) + TB - 1) / TB))

  // zero-init: [m1,s1,agg1] contiguous, [m2,s2] contiguous, d_out
  long z1 = (long)N * (8 + 8 + 128);
  fill_zero_kernel<<<GRID(z1), TB, 0, stream>>>((float*)m1, z1);
  long z2 = (long)N * 2;
  fill_zero_kernel<<<GRID(z2), TB, 0, stream>>>((float*)m2, z2);
  long z3 = (long)N * OUTC;
  fill_zero_kernel<<<GRID(z3), TB, 0, stream>>>(out, z3);

  // ---- layer 1 ----
  {
    int waves = (N / 16) * (128 / 16);                 // 25000
    wmma_gemm_f32_kernel<<<(waves * 32 + TB - 1) / TB, TB, 0, stream>>>(
        x, W1, h1, N, IN_CH, HEADS * HID);
    att_kernel<<<GRID((long)N * HEADS), TB, 0, stream>>>(
        h1, as1, ad1, a1s, a1d, N, HEADS, HID);
    edge_max_kernel<<<GRID(ET), TB, 0, stream>>>(ei, E, N, a1s, a1d, m1, HEADS);
    edge_agg_kernel<<<GRID((long)ET * HEADS), TB, 0, stream>>>(
        ei, E, N, a1s, a1d, m1, s1, h1, agg1, HEADS, HID);
    norm_kernel<<<GRID((long)N * 128), TB, 0, stream>>>(
        agg1, s1, b1, N, HEADS, HID, /*elu=*/1);
  }

  // ---- layer 2 (1 head, mean over 1 head == identity) ----
  {
    int waves = (N / 16) * (OUTC / 16);                // 3125
    wmma_gemm_f32_kernel<<<(waves * 32 + TB - 1) / TB, TB, 0, stream>>>(
        agg1, W2, h2, N, HEADS * HID, OUTC);
    att_kernel<<<GRID((long)N), TB, 0, stream>>>(
        h2, as2, ad2, a2s, a2d, N, 1, OUTC);
    edge_max_kernel<<<GRID(ET), TB, 0, stream>>>(ei, E, N, a2s, a2d, m2, 1);
    edge_agg_kernel<<<GRID((long)ET), TB, 0, stream>>>(
        ei, E, N, a2s, a2d, m2, s2, h2, out, 1, OUTC);
    norm_kernel<<<GRID((long)N * OUTC), TB, 0, stream>>>(
        out, s2, b2, N, 1, OUTC, /*elu=*/0);
  }
  #undef GRID
}